// DebiasLoss_18081812316254
// MI455X (gfx1250) — compile-verified
//
#include <hip/hip_runtime.h>
#include <hip/hip_bf16.h>
#include <cstdint>

// ---------------------------------------------------------------------------
// DebiasLoss on MI455X (gfx1250): memory-bound streaming softmax reduction.
// 65.5 MB of logits read once -> HBM-bound (~2.8us @ 23.3 TB/s floor; L2-
// resident on replays, 192MB L2). Data path: CDNA5 async global->LDS copies
// (ASYNCcnt-tracked), double-buffered per wave, pass1 from LDS into registers,
// pass2 from registers. Deterministic two-level reduction (no atomics).
// ---------------------------------------------------------------------------

#define ASYNC_LOAD_B128(lds_off_u32, gaddr_u64)                                 \
  asm volatile("global_load_async_to_lds_b128 %0, %1, off"                      \
               :: "v"(lds_off_u32), "v"(gaddr_u64) : "memory")

#define S_WAIT_ASYNCCNT_8() asm volatile("s_wait_asynccnt 8" ::: "memory")
#define S_WAIT_ASYNCCNT_0() asm volatile("s_wait_asynccnt 0" ::: "memory")

namespace {
constexpr int   C        = 1000;   // classes
constexpr int   C4       = 250;    // float4 slots per row (1000 % 4 == 0)
constexpr int   THREADS  = 128;    // 4 wave32 per block
constexpr int   WAVES    = THREADS / 32;
constexpr int   NBLK     = 1024;   // partials in d_ws
constexpr float BETA     = 0.5f;
constexpr float LAMDA    = 1.0f;   // margin_lf scale
}

__global__ __launch_bounds__(THREADS)
void debias_loss_main(const float* __restrict__ logits,
                      const int*   __restrict__ targets,
                      const float* __restrict__ coefp,
                      const float* __restrict__ w_norm,
                      const float* __restrict__ class_bias,
                      float*       __restrict__ partials,
                      int B)
{
  __shared__ __align__(16) float s_ml[C];               // margin_lf
  __shared__ __align__(16) float s_rows[WAVES][2][C];   // double-buffered rows
  __shared__ float s_wacc[WAVES];

  const int tid  = threadIdx.x;
  const int lane = tid & 31;
  const int wave = tid >> 5;

  // margin_lf[c] = log(class_bias[c] + 1e-12), shared by all waves in block
  for (int c = tid; c < C; c += THREADS)
    s_ml[c] = __logf(class_bias[c] + 1e-12f);
  __syncthreads();

  // Hoist this lane's margin_lf slots into registers (loop-invariant).
  const float4* ml4 = (const float4*)s_ml;
  float4 mlr[8];
#pragma unroll
  for (int k = 0; k < 8; ++k) {
    const int s = lane + 32 * k;
    if (s < C4) mlr[k] = ml4[s];
    else        mlr[k] = float4{0.f, 0.f, 0.f, 0.f};
  }

  const float coef       = coefp[0];
  const int   totalWaves = (gridDim.x * blockDim.x) >> 5;
  const int   gwave      = (blockIdx.x * blockDim.x + tid) >> 5;

  // Issue one row (4000 B) as 8 async B128 ops: 7 full + 1 with 26 lanes.
  auto issue_row = [&](int r, int buf) {
    const uint64_t gbase = (uint64_t)(uintptr_t)(logits + (size_t)r * C);
    const uint32_t lbase = (uint32_t)(uintptr_t)(&s_rows[wave][buf][0]);
#pragma unroll
    for (int k = 0; k < 7; ++k) {
      uint64_t ga = gbase + (uint64_t)(lane * 16 + k * 512);
      uint32_t la = lbase + (uint32_t)(lane * 16 + k * 512);
      ASYNC_LOAD_B128(la, ga);
    }
    if (lane < C4 - 7 * 32) {  // 26 tail lanes
      uint64_t ga = gbase + (uint64_t)(lane * 16 + 7 * 512);
      uint32_t la = lbase + (uint32_t)(lane * 16 + 7 * 512);
      ASYNC_LOAD_B128(la, ga);
    }
  };

  float acc = 0.0f;
  int buf = 0;
  if (gwave < B) issue_row(gwave, 0);

  for (int r = gwave; r < B; r += totalWaves) {
    const int nxt = r + totalWaves;
    if (nxt < B) { issue_row(nxt, buf ^ 1); S_WAIT_ASYNCCNT_8(); }
    else         { S_WAIT_ASYNCCNT_0(); }

    const float4* v4 = (const float4*)&s_rows[wave][buf][0];

    const int   t   = targets[r];
    const float tgt = logits[(size_t)r * C + t];  // uniform scalar gather
    const float wn  = w_norm[t];
    const float mlt = s_ml[t];

    // Pass 1: LDS -> registers, row max of adj (target col = -inf) + keep
    float4 x[8];
    float m = -INFINITY;
    int keep = 0;
#pragma unroll
    for (int k = 0; k < 8; ++k) {
      const int s = lane + 32 * k;
      if (s < C4) {
        const float4 v = v4[s];
        const float4 g = mlr[k];
        const int cb = 4 * s;
        x[k].x = (cb + 0 == t) ? -INFINITY : v.x + g.x;
        x[k].y = (cb + 1 == t) ? -INFINITY : v.y + g.y;
        x[k].z = (cb + 2 == t) ? -INFINITY : v.z + g.z;
        x[k].w = (cb + 3 == t) ? -INFINITY : v.w + g.w;
        keep |= (v.x > tgt) | (v.y > tgt) | (v.z > tgt) | (v.w > tgt);
        m = fmaxf(m, fmaxf(fmaxf(x[k].x, x[k].y), fmaxf(x[k].z, x[k].w)));
      } else {
        x[k] = float4{-INFINITY, -INFINITY, -INFINITY, -INFINITY};
      }
    }
#pragma unroll
    for (int o = 16; o > 0; o >>= 1) m = fmaxf(m, __shfl_xor(m, o, 32));

    // Pass 2: registers only; exp(-inf)=0 covers target col and tail slots
    float sum = 0.0f;
#pragma unroll
    for (int k = 0; k < 8; ++k) {
      sum += __expf(x[k].x - m) + __expf(x[k].y - m)
           + __expf(x[k].z - m) + __expf(x[k].w - m);
    }
#pragma unroll
    for (int o = 16; o > 0; o >>= 1) sum += __shfl_xor(sum, o, 32);

    const int keepAny = __any(keep);

    // Fold target column back in analytically
    float d = 0.0f;
    if (keepAny) {
      const float q = tgt / wn - wn;
      d = BETA * coef * log1pf(q * q);
    }
    const float at = tgt + LAMDA * mlt - d;
    const float M2 = fmaxf(m, at);
    const float S  = sum * __expf(m - M2) + __expf(at - M2);
    acc += __logf(S) + M2 - at;   // logsumexp(adj) - adj[target]

    buf ^= 1;
  }

  if (lane == 0) s_wacc[wave] = acc;
  __syncthreads();
  if (tid == 0) {
    float b = 0.0f;
#pragma unroll
    for (int w = 0; w < WAVES; ++w) b += s_wacc[w];
    partials[blockIdx.x] = b;
  }
}

__global__ __launch_bounds__(256)
void debias_loss_finalize(const float* __restrict__ partials,
                          float* __restrict__ out, int n, float invB)
{
  __shared__ float sh[256];
  float v = 0.0f;
  for (int i = threadIdx.x; i < n; i += 256) v += partials[i];
  sh[threadIdx.x] = v;
  __syncthreads();
  for (int s = 128; s > 0; s >>= 1) {
    if (threadIdx.x < s) sh[threadIdx.x] += sh[threadIdx.x + s];
    __syncthreads();
  }
  if (threadIdx.x == 0) out[0] = sh[0] * invB;
}

extern "C" void kernel_launch(void* const* d_in, const int* in_sizes, int n_in,
                              void* d_out, int out_size, void* d_ws, size_t ws_size,
                              hipStream_t stream) {
  const float* logits     = (const float*)d_in[0];
  const int*   targets    = (const int*)d_in[1];
  const float* coef       = (const float*)d_in[2];
  const float* w_norm     = (const float*)d_in[3];
  const float* class_bias = (const float*)d_in[4];
  const int B = in_sizes[1];          // targets length = batch

  float* partials = (float*)d_ws;     // NBLK floats, fully rewritten each call

  debias_loss_main<<<NBLK, THREADS, 0, stream>>>(
      logits, targets, coef, w_norm, class_bias, partials, B);
  debias_loss_finalize<<<1, 256, 0, stream>>>(
      partials, (float*)d_out, NBLK, 1.0f / (float)B);
}